// MultiHeadGATLayer_55009941127939
// MI455X (gfx1250) — compile-verified
//
#include <hip/hip_runtime.h>
#include <hip/hip_bf16.h>
#include <math.h>

typedef __attribute__((ext_vector_type(2))) float v2f;
typedef __attribute__((ext_vector_type(8))) float v8f;

#define N_NODES 50000
#define DEG     16
#define IN_DIM  128
#define OUT_DIM 64
#define HEADS   4

// -------------------------------------------------------------------------
// Kernel 1: z[h] = h @ fc_w[h]^T  via V_WMMA_F32_16X16X4_F32 (f32 WMMA).
// One wave computes one 16x16 C tile. 3125 Mtiles * 4 Ntiles * 4 heads =
// 50000 tiles; launched so every wave has a full tile (EXEC all-1s).
// A-frag layout (16x4 f32): lanes 0-15 -> M=lane, K={0,1}; lanes 16-31 ->
// M=lane-16, K={2,3}. B-frag (4x16) mirrored. C/D: lanes0-15 M=v, lanes16-31
// M=v+8, N=lane&15.
// -------------------------------------------------------------------------
__global__ __launch_bounds__(256) void gat_gemm_wmma(
    const float* __restrict__ hmat,   // [N, 128]
    const float* __restrict__ fcw,    // [HEADS, 64, 128]
    float* __restrict__ z)            // [HEADS, N, 64]
{
    const int wave = blockIdx.x * (blockDim.x >> 5) + (threadIdx.x >> 5);
    const int lane = threadIdx.x & 31;

    const int head = wave & 3;          // 4 heads
    const int ct   = (wave >> 2) & 3;   // 4 column tiles of 16 (OUT_DIM=64)
    const int mt   = wave >> 4;         // 3125 row tiles of 16
    if (mt >= N_NODES / 16) return;     // wave-uniform; never diverges

    const int mrow  = mt * 16 + (lane & 15);
    const int ncol  = ct * 16 + (lane & 15);
    const int khalf = (lane >> 4) ? 2 : 0;

    const float* __restrict__ A = hmat + (size_t)mrow * IN_DIM;
    const float* __restrict__ B = fcw + ((size_t)head * OUT_DIM + ncol) * IN_DIM;

    v8f c = {};
#pragma unroll 8
    for (int kk = 0; kk < IN_DIM; kk += 4) {
        v2f a, b;
        a.x = A[kk + khalf];
        a.y = A[kk + khalf + 1];
        b.x = B[kk + khalf];
        b.y = B[kk + khalf + 1];
        c = __builtin_amdgcn_wmma_f32_16x16x4_f32(
                /*neg_a=*/false, a, /*neg_b=*/false, b,
                /*c_mod=*/(short)0, c, /*reuse_a=*/false, /*reuse_b=*/false);
    }

    const int mbase = mt * 16 + ((lane >> 4) ? 8 : 0);
    float* __restrict__ Z = z + (size_t)head * N_NODES * OUT_DIM;
#pragma unroll
    for (int v = 0; v < 8; ++v) {
        Z[(size_t)(mbase + v) * OUT_DIM + ct * 16 + (lane & 15)] = c[v];
    }
}

// -------------------------------------------------------------------------
// Kernel 2: per-node attention. One wave32 per node; lane owns output dims
// {2*lane, 2*lane+1}. entmax-1.5 over the 16 neighbors is done entirely in
// registers: logits one-per-lane (lanes 0..15), bitonic sort via shfl_xor,
// prefix sums via shfl_up, support size via ballot.
// -------------------------------------------------------------------------
__device__ __forceinline__ float wave_sum(float v) {
#pragma unroll
    for (int m = 16; m >= 1; m >>= 1) v += __shfl_xor(v, m, 32);
    return v;
}

__global__ __launch_bounds__(256) void gat_attn(
    const float* __restrict__ z,        // [HEADS, N, 64]
    const int*   __restrict__ src,      // [N*16]
    const float* __restrict__ we,       // [N*16]
    const float* __restrict__ attn_w,   // [HEADS, 128]
    const float* __restrict__ head_w,   // [HEADS]
    float*       __restrict__ out)      // [N, 64]
{
    const int n    = blockIdx.x * (blockDim.x >> 5) + (threadIdx.x >> 5);
    const int lane = threadIdx.x & 31;
    if (n >= N_NODES) return;

    // softmax over the 4 head weights (redundant per-lane scalar work)
    float hw0 = head_w[0], hw1 = head_w[1], hw2 = head_w[2], hw3 = head_w[3];
    float hm  = fmaxf(fmaxf(hw0, hw1), fmaxf(hw2, hw3));
    float ex0 = __expf(hw0 - hm), ex1 = __expf(hw1 - hm);
    float ex2 = __expf(hw2 - hm), ex3 = __expf(hw3 - hm);
    float inv = 1.0f / (ex0 + ex1 + ex2 + ex3);
    float ws[HEADS] = {ex0 * inv, ex1 * inv, ex2 * inv, ex3 * inv};

    const int d0 = 2 * lane;

    // lane j<16 owns edge j's neighbor index and edge weight
    int   myidx = (lane < DEG) ? src[n * DEG + lane] : 0;
    float myw   = (lane < DEG) ? we[n * DEG + lane] : 0.0f;

    float acc0 = 0.0f, acc1 = 0.0f;

#pragma unroll
    for (int head = 0; head < HEADS; ++head) {
        const float* __restrict__ Zh = z + (size_t)head * N_NODES * OUT_DIM;
        const float* __restrict__ aw = attn_w + head * 2 * OUT_DIM;

        const float as0 = aw[d0],           as1 = aw[d0 + 1];
        const float ad0 = aw[OUT_DIM + d0], ad1 = aw[OUT_DIM + d0 + 1];

        // destination term: z[n] . a_dst
        const float2 zd = *(const float2*)(Zh + (size_t)n * OUT_DIM + d0);
        const float dotdst = wave_sum(zd.x * ad0 + zd.y * ad1);

        float r0[DEG], r1[DEG];
        float e_mine = 0.0f;
#pragma unroll
        for (int j = 0; j < DEG; ++j) {
            const int s = __shfl(myidx, j, 32);
            const float2 v = *(const float2*)(Zh + (size_t)s * OUT_DIM + d0);
            r0[j] = v.x; r1[j] = v.y;
            float e = wave_sum(v.x * as0 + v.y * as1) + dotdst;
            e = (e > 0.0f) ? e : 0.01f * e;          // leaky_relu
            if (lane == j) e_mine = e + myw;
        }

        // ---------------- entmax-1.5 over lanes 0..15 ----------------
        float zhalf = e_mine * 0.5f;
        float mval  = (lane < DEG) ? zhalf : -INFINITY;
#pragma unroll
        for (int m = 16; m >= 1; m >>= 1) mval = fmaxf(mval, __shfl_xor(mval, m, 32));
        const float z0 = zhalf - mval;               // original edge order (edge = lane)

        // bitonic sort descending across 32 lanes; lanes>=16 padded -inf
        float v = (lane < DEG) ? z0 : -INFINITY;
#pragma unroll
        for (int k = 2; k <= 32; k <<= 1) {
            const bool desc = ((lane & k) == 0);
#pragma unroll
            for (int j = k >> 1; j >= 1; j >>= 1) {
                const float o  = __shfl_xor(v, j, 32);
                const bool low = ((lane & j) == 0);
                const float mx = fmaxf(v, o), mn = fminf(v, o);
                v = desc ? (low ? mx : mn) : (low ? mn : mx);
            }
        }
        // v = zs (i-th largest in lane i). Inclusive prefix sums.
        float cs = v, cs2 = v * v;
#pragma unroll
        for (int off = 1; off < 32; off <<= 1) {
            const float t  = __shfl_up(cs,  off, 32);
            const float t2 = __shfl_up(cs2, off, 32);
            if (lane >= off) { cs += t; cs2 += t2; }
        }
        const float kf     = (float)(lane + 1);
        const float mean   = cs / kf;
        const float meansq = cs2 / kf;
        const float ssv    = kf * (meansq - mean * mean);
        const float delta  = fmaxf((1.0f - ssv) / kf, 0.0f);
        const float tau    = mean - sqrtf(delta);

        const unsigned long long bal = __ballot((lane < DEG) && (tau <= v));
        const int support = __popcll(bal & 0xFFFFull);
        const float tau_star = __shfl(tau, support - 1, 32);

        float p = fmaxf(z0 - tau_star, 0.0f);
        const float alpha = p * p;                   // lane j holds alpha for edge j

        // out_n += ws[head] * sum_j alpha_j * zsrc_j
        const float wsh = ws[head];
#pragma unroll
        for (int j = 0; j < DEG; ++j) {
            const float aj = __shfl(alpha, j, 32) * wsh;
            acc0 = fmaf(aj, r0[j], acc0);
            acc1 = fmaf(aj, r1[j], acc1);
        }
    }

    float2* O = (float2*)(out + (size_t)n * OUT_DIM + d0);
    *O = make_float2(acc0, acc1);
}

// -------------------------------------------------------------------------
extern "C" void kernel_launch(void* const* d_in, const int* in_sizes, int n_in,
                              void* d_out, int out_size, void* d_ws, size_t ws_size,
                              hipStream_t stream) {
    const float* hmat   = (const float*)d_in[0];   // [N,128]
    const int*   src    = (const int*)  d_in[1];   // [E]
    const float* we     = (const float*)d_in[2];   // [E,1]
    const float* fcw    = (const float*)d_in[3];   // [4,64,128]
    const float* attn_w = (const float*)d_in[4];   // [4,128]
    const float* head_w = (const float*)d_in[5];   // [4]
    float* out = (float*)d_out;
    float* z   = (float*)d_ws;                     // [4, N, 64] = 51.2 MB

    // Kernel 1: 3125 Mtiles * 4 Ntiles * 4 heads = 50000 tiles, 8 waves/block
    const int tiles  = (N_NODES / 16) * 4 * HEADS;     // 50000
    gat_gemm_wmma<<<tiles / 8, 256, 0, stream>>>(hmat, fcw, z);

    // Kernel 2: one wave per node, 8 waves/block
    gat_attn<<<(N_NODES + 7) / 8, 256, 0, stream>>>(z, src, we, attn_w, head_w, out);
}